// Ballistics_62534723830254
// MI455X (gfx1250) — compile-verified
//
#include <hip/hip_runtime.h>
#include <math.h>

// Chunked contractive-scan parameters.
//  - CHUNK_L: output samples produced per thread (start offsets are multiples
//             of 512 -> float4 / b128 aligned)
//  - Warm-up W is ADAPTIVE per row: the step map y -> y + c*(x-y) contracts
//    with slope (1-c) < 1, so after W steps the influence of the guessed
//    initial state is <= (1-c_min)^W <= e^{-W*c_min}. W = 20/c_min gives a
//    residual ~2e-9 (below fp32 eps for O(1) states), clamped to [32, 4096].
#define CHUNK_L   512
#define WARM_MAX  4096
#define WARM_MIN  32

// Async-to-LDS staging: each lane double-buffers 16-float tiles of its own
// input stream through LDS (4 x b128 per tile). Padded row stride (5 float4 =
// 20 dwords, gcd(20,64)=4) keeps ds_load_b128 at worst 2-way bank conflicted.
#define TILE  16
#define LROW  5            // float4 stride per lane row (4 data + 1 pad)
#define NLANE 256          // block size

#if defined(__HIP_DEVICE_COMPILE__) && \
    __has_builtin(__builtin_amdgcn_global_load_async_to_lds_b128)
#define USE_ASYNC_LDS 1
#else
#define USE_ASYNC_LDS 0
#endif

#if USE_ASYNC_LDS
// Builtin parameter pointee type (from clang diagnostic): int __vector_size__(16).
// HIP allows implicit generic <-> __device__/__shared__ addrspace conversion,
// so plain generic pointers of the right pointee type are accepted.
typedef int v4i_async __attribute__((vector_size(16)));
#define TO_GLOBAL(p) ((v4i_async*)(p))
#define TO_LOCAL(p)  ((v4i_async*)(p))
#if __has_builtin(__builtin_amdgcn_s_wait_asynccnt)
#define WAIT_ASYNC(n) __builtin_amdgcn_s_wait_asynccnt(n)
#else
#define WAIT_ASYNC(n) asm volatile("s_wait_asynccnt %0" :: "i"(n) : "memory")
#endif
#endif

__global__ __launch_bounds__(NLANE) void Ballistics_scan_kernel(
    const float* __restrict__ signal,   // [B][T]
    const float* __restrict__ z_alpha,  // [B][2]
    float* __restrict__ out,            // [B][T]
    int B, int T, int numChunks)
{
    int tid   = blockIdx.x * blockDim.x + threadIdx.x;
    int total = B * numChunks;
    if (tid >= total) return;

    int b = tid / numChunks;
    int c = tid - b * numChunks;

    // Per-row coefficients: at = sigmoid(z[0]), rt = sigmoid(z[1])
    float za = z_alpha[2 * b + 0];
    float zr = z_alpha[2 * b + 1];
    float at = 1.0f / (1.0f + __expf(-za));
    float rt = 1.0f / (1.0f + __expf(-zr));

    // Adaptive warm-up length from the row's contraction rate.
    float cmin = fminf(at, rt);
    int   W    = (int)ceilf(20.0f / cmin);
    W = (W < WARM_MIN) ? WARM_MIN : (W > WARM_MAX ? WARM_MAX : W);

    const float* __restrict__ row  = signal + (long long)b * (long long)T;
    float*       __restrict__ orow = out    + (long long)b * (long long)T;

    int start = c * CHUNK_L;
    int end   = start + CHUNK_L;
    if (end > T) end = T;

    int   warm = (start - W) & ~3;    // float4-aligned warm start
    float y;
    if (warm <= 0) {
        warm = 0;
        y = 1.0f;                     // exact initial state (zi = ones)
    } else {
        y = row[warm];                // arbitrary guess; contraction erases it
    }

    // One recurrence step: 3-op dependent chain (v_cmp||v_sub -> cndmask -> fma)
#define STEP(xv)  do { float _x = (xv);                       \
                       float _cf = (_x > y) ? at : rt;        \
                       y = fmaf(_cf, _x - y, y); } while (0)

    // ---- warm-up: converge to the true trajectory, no output ----
    for (int t = warm; t < start; t += 4) {
        if ((t & 31) == 0) {
            int pf = t + 512; if (pf >= T) pf = T - 1;
            __builtin_prefetch((const void*)(row + pf), 0, 0);  // global_prefetch_b8
        }
        float4 x4 = *(const float4*)(row + t);                  // global_load_b128
        STEP(x4.x); STEP(x4.y); STEP(x4.z); STEP(x4.w);
    }

    int t = start;

#if USE_ASYNC_LDS
    // ---- main: double-buffered async-to-LDS staged tiles ----
    __shared__ float4 smem[2 * NLANE * LROW];     // 40 KB
    float4* lrow0 = &smem[threadIdx.x * LROW];    // this lane's row, buffer 0

    auto issue_tile = [&](int tg, int buf) {
        const float4* g = (const float4*)(row + tg);
        float4*       l = lrow0 + buf * (NLANE * LROW);
#pragma unroll
        for (int j = 0; j < TILE / 4; ++j)
            __builtin_amdgcn_global_load_async_to_lds_b128(
                TO_GLOBAL(g + j), TO_LOCAL(l + j), 0, 0);
    };

    int nT = (end - t) / TILE;
    if (nT > 0) {
        issue_tile(t, 0);
        for (int k = 0; k < nT; ++k) {
            int tcur = t + k * TILE;
            __builtin_prefetch((const void*)(row + ((tcur + 512 < T) ? tcur + 512
                                                                     : T - 1)), 0, 0);
            bool more = (k + 1 < nT);
            if (more) issue_tile(tcur + TILE, (k + 1) & 1);
            // In-order completion: <= 4 outstanding => tile k has landed.
            if (more) { WAIT_ASYNC(TILE / 4); } else { WAIT_ASYNC(0); }
            asm volatile("" ::: "memory");

            float4* l = lrow0 + (k & 1) * (NLANE * LROW);
#pragma unroll
            for (int j = 0; j < TILE / 4; ++j) {
                float4 x4 = l[j];                       // ds_load_b128
                float4 o4;
                STEP(x4.x); o4.x = y;
                STEP(x4.y); o4.y = y;
                STEP(x4.z); o4.z = y;
                STEP(x4.w); o4.w = y;
                *(float4*)(orow + tcur + j * 4) = o4;   // global_store_b128
            }
        }
        t += nT * TILE;
    }
#endif

    // ---- remainder (and fallback main loop): direct b128 path ----
    for (; t + 3 < end; t += 4) {
        if ((t & 31) == 0) {
            int pf = t + 512; if (pf >= T) pf = T - 1;
            __builtin_prefetch((const void*)(row + pf), 0, 0);
        }
        float4 x4 = *(const float4*)(row + t);
        float4 o4;
        STEP(x4.x); o4.x = y;
        STEP(x4.y); o4.y = y;
        STEP(x4.z); o4.z = y;
        STEP(x4.w); o4.w = y;
        *(float4*)(orow + t) = o4;
    }
    for (; t < end; ++t) {            // scalar tail (only if T % 4 != 0)
        float x = row[t];
        STEP(x);
        orow[t] = y;
    }
#undef STEP
}

extern "C" void kernel_launch(void* const* d_in, const int* in_sizes, int n_in,
                              void* d_out, int out_size, void* d_ws, size_t ws_size,
                              hipStream_t stream)
{
    const float* signal  = (const float*)d_in[0];   // [B][T] f32
    const float* z_alpha = (const float*)d_in[1];   // [B][2] f32
    float*       out     = (float*)d_out;           // [B][T] f32

    int B = in_sizes[1] / 2;
    int T = in_sizes[0] / B;

    int numChunks = (T + CHUNK_L - 1) / CHUNK_L;
    int total     = B * numChunks;                  // 64 * 512 = 32768 threads
    int block     = NLANE;                          // 8 wave32 waves / WG
    int grid      = (total + block - 1) / block;

    Ballistics_scan_kernel<<<grid, block, 0, stream>>>(signal, z_alpha, out,
                                                       B, T, numChunks);
}